// WindowedSelfRelativeAttention_34703335751907
// MI455X (gfx1250) — compile-verified
//
#include <hip/hip_runtime.h>

typedef float v2f __attribute__((ext_vector_type(2)));
typedef float v8f __attribute__((ext_vector_type(8)));

// D = A(16x4, f32) * B(4x16, f32) + C(16x16, f32), exact f32 — CDNA5 WMMA.
#define WMMA_F32X4(a, b, c) \
  __builtin_amdgcn_wmma_f32_16x16x4_f32(false, (a), false, (b), (short)0, (c), false, false)

constexpr int Bsz = 8, H = 16, S = 2048, D = 64;
constexpr int W = 11;      // window offsets -5..+5
constexpr int TS = 16;     // rows (s positions) per tile / wave
constexpr int BAND = 32;   // staged K/V band columns: s0-8 .. s0+23
constexpr int ST = 68;     // LDS row stride (floats): 272B, 16B-aligned rows,
                           // banks (4*ln+d)&63 -> conflict-free lane gathers

// Async global -> LDS, 16 bytes per lane. Tracked with ASYNCcnt.
__device__ __forceinline__ void async_ld_b128(uint32_t lds_byte, uint32_t gbyte_off,
                                              const void* sbase) {
  asm volatile("global_load_async_to_lds_b128 %0, %1, %2"
               :: "v"(lds_byte), "v"(gbyte_off),
                  "s"((unsigned long long)(uintptr_t)sbase)
               : "memory");
}

__device__ __forceinline__ void wait_async0() {
  asm volatile("s_wait_asynccnt 0" ::: "memory");
}

__device__ __forceinline__ uint32_t lds_off(const void* p) {
  // flat shared address carries the LDS byte offset in its low 32 bits
  return (uint32_t)(uintptr_t)p;
}

__global__ __launch_bounds__(32)
void winattn_kernel(const float* __restrict__ q, const float* __restrict__ k,
                    const float* __restrict__ v, const float* __restrict__ ek,
                    const float* __restrict__ ev, float* __restrict__ out) {
  __shared__ float Qt[16][ST];    // q tile          [m][d]
  __shared__ float Kb[32][ST];    // k band          [c][d]
  __shared__ float Vb[32][ST];    // v band          [c][d]
  __shared__ float Ekp[16][ST];   // ek rows (padded)[w][d]
  __shared__ float Evp[16][ST];   // ev rows (padded)[w][d]
  __shared__ float Sf[16][33];    // QK^T band tile  [m][c]
  __shared__ float Rr[16][17];    // Q ek^T tile     [m][w]
  __shared__ float Wb[16][33];    // banded weights  [m][c]
  __shared__ float W16[16][17];   // padded weights  [m][w]

  const int lane = threadIdx.x;        // 0..31, one full wave32
  const int half = lane >> 4;          // 0 | 1
  const int ln   = lane & 15;

  const int tilesPerBH = S / TS;                  // 128
  const int bh = blockIdx.x / tilesPerBH;         // fused b*H+h
  const int s0 = (blockIdx.x % tilesPerBH) * TS;  // tile row base
  const long base = (long)bh * S * D;

  const float* qbh = q + base;
  const float* kbh = k + base;
  const float* vbh = v + base;

  const uint32_t QtB = lds_off(&Qt[0][0]);
  const uint32_t KbB = lds_off(&Kb[0][0]);
  const uint32_t VbB = lds_off(&Vb[0][0]);
  const uint32_t EkB = lds_off(&Ekp[0][0]);
  const uint32_t EvB = lds_off(&Evp[0][0]);

  // ---- async-stage tiles into LDS, 16B (4-float) chunks per lane ----
  // Qt: 16 rows * 16 chunks = 256 chunks
#pragma unroll
  for (int it = 0; it < 8; ++it) {
    const int cid = it * 32 + lane;
    const int r = cid >> 4, d4 = cid & 15;
    async_ld_b128(QtB + (uint32_t)(r * ST + d4 * 4) * 4u,
                  (uint32_t)(((s0 + r) * D + d4 * 4) * 4), qbh);
  }
  // K/V band: 32 rows * 16 chunks = 512 chunks each
#pragma unroll
  for (int it = 0; it < 16; ++it) {
    const int cid = it * 32 + lane;
    const int c = cid >> 4, d4 = cid & 15;
    const int gs = (s0 - 8 + c + S) & (S - 1);     // circular roll, S pow2
    const uint32_t go = (uint32_t)((gs * D + d4 * 4) * 4);
    async_ld_b128(KbB + (uint32_t)(c * ST + d4 * 4) * 4u, go, kbh);
    async_ld_b128(VbB + (uint32_t)(c * ST + d4 * 4) * 4u, go, vbh);
  }
  // Embedding rows 0..10 (row = w+11 in the 33-row tables): 176 chunks
#pragma unroll
  for (int it = 0; it < 6; ++it) {
    const int cid = it * 32 + lane;
    if (cid < W * 16) {
      const int w = cid >> 4, d4 = cid & 15;
      const uint32_t go = (uint32_t)(((w + 11) * D + d4 * 4) * 4);
      async_ld_b128(EkB + (uint32_t)(w * ST + d4 * 4) * 4u, go, ek);
      async_ld_b128(EvB + (uint32_t)(w * ST + d4 * 4) * 4u, go, ev);
    }
  }
  // zero-pad emb rows 11..15 (read by the padded-K WMMA steps)
  for (int idx = lane; idx < 5 * D; idx += 32) {
    const int w = W + (idx >> 6), d = idx & 63;
    Ekp[w][d] = 0.0f;
    Evp[w][d] = 0.0f;
  }
  wait_async0();
  __syncthreads();

  // ---- scores: Sf = Q * Kband^T (16x32), Rr = Q * ek^T (16x16) ----
  // A layout: lane=M(ln), K = 2*half + vgpr.  B layout: lane=N(ln), same K striping.
  v8f accS0 = {}, accS1 = {}, accR = {};
#pragma unroll
  for (int kk = 0; kk < 16; ++kk) {
    const int d0 = 4 * kk + 2 * half;
    v2f a;  a.x  = Qt[ln][d0];        a.y  = Qt[ln][d0 + 1];
    v2f b0; b0.x = Kb[ln][d0];        b0.y = Kb[ln][d0 + 1];
    v2f b1; b1.x = Kb[16 + ln][d0];   b1.y = Kb[16 + ln][d0 + 1];
    v2f b2; b2.x = Ekp[ln][d0];       b2.y = Ekp[ln][d0 + 1];
    accS0 = WMMA_F32X4(a, b0, accS0);
    accS1 = WMMA_F32X4(a, b1, accS1);
    accR  = WMMA_F32X4(a, b2, accR);
  }
  // C/D layout: vgpr j -> M = j + 8*half, N = ln
#pragma unroll
  for (int j = 0; j < 8; ++j) {
    const int m = j + 8 * half;
    Sf[m][ln]      = accS0[j];
    Sf[m][16 + ln] = accS1[j];
    Rr[m][ln]      = accR[j];
  }
  __syncthreads();

  // ---- per-row softmax over the 11 window offsets; build weight matrices ----
  if (lane < 16) {
    const int m = lane;
    float sc[W];
    float mx = -1e30f;
#pragma unroll
    for (int w = 0; w < W; ++w) {
      // band column for (row m, offset w-5): c = m + w + 3
      sc[w] = 0.125f * (Sf[m][m + w + 3] + Rr[m][w]);   // scale = 1/sqrt(64)
      mx = fmaxf(mx, sc[w]);
    }
    float sum = 0.0f;
#pragma unroll
    for (int w = 0; w < W; ++w) { sc[w] = __expf(sc[w] - mx); sum += sc[w]; }
    const float inv = 1.0f / sum;
#pragma unroll
    for (int c = 0; c < 32; ++c) Wb[m][c] = 0.0f;
#pragma unroll
    for (int w = 0; w < W; ++w) Wb[m][m + w + 3] = sc[w] * inv;
#pragma unroll
    for (int w = 0; w < 16; ++w) W16[m][w] = (w < W) ? sc[w] * inv : 0.0f;
  }
  __syncthreads();

  // ---- out = Wb(16x32) * Vband(32x64)  +  W16(16x16) * ev(16x64) ----
#pragma unroll
  for (int nt = 0; nt < 4; ++nt) {
    v8f acc = {};
#pragma unroll
    for (int kk = 0; kk < 8; ++kk) {           // K over 32 band columns
      const int c0 = 4 * kk + 2 * half;
      v2f a; a.x = Wb[ln][c0];             a.y = Wb[ln][c0 + 1];
      v2f b; b.x = Vb[c0][16 * nt + ln];   b.y = Vb[c0 + 1][16 * nt + ln];
      acc = WMMA_F32X4(a, b, acc);
    }
#pragma unroll
    for (int kk = 0; kk < 4; ++kk) {           // K over 16 (11 valid) offsets
      const int w0 = 4 * kk + 2 * half;
      v2f a; a.x = W16[ln][w0];            a.y = W16[ln][w0 + 1];
      v2f b; b.x = Evp[w0][16 * nt + ln];  b.y = Evp[w0 + 1][16 * nt + ln];
      acc = WMMA_F32X4(a, b, acc);
    }
#pragma unroll
    for (int j = 0; j < 8; ++j) {
      const int m = j + 8 * half;
      out[base + (long)(s0 + m) * D + 16 * nt + ln] = acc[j];
    }
  }
}

extern "C" void kernel_launch(void* const* d_in, const int* in_sizes, int n_in,
                              void* d_out, int out_size, void* d_ws, size_t ws_size,
                              hipStream_t stream) {
  const float* q  = (const float*)d_in[0];
  const float* k  = (const float*)d_in[1];
  const float* v  = (const float*)d_in[2];
  const float* ek = (const float*)d_in[3];
  const float* ev = (const float*)d_in[4];
  float* out = (float*)d_out;

  dim3 grid(Bsz * H * (S / TS));   // 16384 tiles, one wave32 each
  dim3 block(32);
  hipLaunchKernelGGL(winattn_kernel, grid, block, 0, stream, q, k, v, ek, ev, out);
}